// Fourdloss_16621523436335
// MI455X (gfx1250) — compile-verified
//
#include <hip/hip_runtime.h>
#include <hip/hip_bf16.h>
#include <math.h>

typedef float v2f __attribute__((ext_vector_type(2)));
typedef float v8f __attribute__((ext_vector_type(8)));
typedef unsigned int u32x4 __attribute__((ext_vector_type(4)));
typedef unsigned int u32x8 __attribute__((ext_vector_type(8)));

#define N_TOT 8192
#define BSZ   4096
#define DIM   128
#define PANEL 64          // columns staged in LDS per pipeline stage
#define LDS_STRIDE 130    // 128 + 2 pad floats -> conflict-free b64 LDS reads
#define PANEL_FLOATS (PANEL * LDS_STRIDE)          // floats per buffer
#define PANEL_BYTES  (PANEL_FLOATS * 4)            // 33280 B per buffer
#define NPANEL (N_TOT / PANEL)
#define WAVES_PER_BLOCK 8
#define ROWS_PER_BLOCK (WAVES_PER_BLOCK * 16)
#define INV_TEMP 20.0f    // 1 / 0.05
#define EPSF 1e-6f

// cf[row][k] = features[row % 4096][row / 4096][k]; returns float offset of row start
__device__ __forceinline__ int cf_base(int row) {
    return (((row & (BSZ - 1)) << 1) + (row >> 12)) * DIM;
}

__global__ void init_ws_kernel(float* ws) {
    ws[threadIdx.x] = 0.0f;   // zero S accumulator + 128 histogram bins (+ slack)
}

__global__ void hist_kernel(const int* __restrict__ labels, unsigned* __restrict__ hist) {
    int i = blockIdx.x * blockDim.x + threadIdx.x;
    if (i < BSZ) atomicAdd(&hist[labels[i] & 127], 1u);
}

// Issue a TDM load of one panel (PANEL cf-columns x DIM fp32) into LDS.
// 2D tile: tile_dim0=128 elems/row, tile_dim1=64 rows, row stride 256 elems
// (consecutive cf columns are 2 feature rows apart within a view).
// LDS padding: +2 DWORDs after every 128 DWORDs -> matches LDS_STRIDE=130.
__device__ __forceinline__ void tdm_load_panel(const float* gbase, unsigned lds_byte_off) {
    unsigned long long ga = (unsigned long long)gbase;
    u32x4 g0;
    g0.x = 1u;                                        // count=1, user descriptor
    g0.y = lds_byte_off;                              // LDS byte address
    g0.z = (unsigned)(ga & 0xFFFFFFFFu);              // global_addr[31:0]
    g0.w = (unsigned)((ga >> 32) & 0x01FFFFFFu)       // global_addr[56:32]
         | (2u << 30);                                // type=2 ("image")
    u32x8 g1;
    g1.s0 = (2u << 16)      // data_size = 4 bytes
          | (1u << 20)      // pad_enable
          | (6u << 22)      // pad_interval: 128 DWORDs
          | (1u << 25);     // pad_amount: 2 DWORDs
    g1.s1 = (128u << 16);   // tensor_dim0 = 128 (atomic_barrier_addr = 0)
    g1.s2 = (0xFFFFu << 16);// tensor_dim1 = 65535 (no OOB within tile)
    g1.s3 = (128u << 16);   // tile_dim0 = 128
    g1.s4 = (unsigned)PANEL;// tile_dim1 = 64, tile_dim2 = 0 (2D)
    g1.s5 = 256u;           // tensor_dim0_stride = 256 elements
    g1.s6 = 0u;
    g1.s7 = 0u;
    u32x4 gz = {0u, 0u, 0u, 0u};
    asm volatile("tensor_load_to_lds %0, %1, %2, %3"
                 :: "s"(g0), "s"(g1), "s"(gz), "s"(gz) : "memory");
}

__device__ __forceinline__ unsigned lds_offset_of(const void* p) {
    // Flat LDS-aperture addresses map to LDS by truncation (ISA 10.2):
    // low 32 bits are the workgroup-relative LDS byte offset.
    return (unsigned)(unsigned long long)p;
}

// Flash-style masked-softmax-sum over the 8192x8192 fp32 Gram matrix.
// TDM double-buffers the column panel; waves run V_WMMA_F32_16X16X4_F32.
__global__ void __launch_bounds__(256, 1)
gram_expsum_kernel(const float* __restrict__ feat,
                   const int* __restrict__ labels,
                   float* __restrict__ Sacc) {
    extern __shared__ float lds[];   // 2 * PANEL_FLOATS

    const int tid  = threadIdx.x;
    const int wave = tid >> 5;
    const int lane = tid & 31;
    const int hf   = lane >> 4;   // half-wave select (0: M=0..7 / K+0, 1: M=8..15 / K+2)
    const int l16  = lane & 15;
    const int kc   = hf << 1;     // K sub-offset for A/B fragments

    const int rowbase = blockIdx.x * ROWS_PER_BLOCK + wave * 16;
    const int arow    = rowbase + l16;

    // ---- Kick off TDM for panel 0 into buffer 0 (overlaps A preload) ----
    if (wave == 0)
        tdm_load_panel(feat + cf_base(0), lds_offset_of(lds));

    // ---- Preload this wave's A fragments for all 32 K-steps (K=128, 4 per WMMA) ----
    v2f areg[32];
    {
        const float* aptr = feat + cf_base(arow) + kc;
        #pragma unroll
        for (int t = 0; t < 32; ++t)
            areg[t] = *(const v2f*)(aptr + 4 * t);
    }

    // Row labels for the diff mask (acc[r] holds row M = r + 8*hf)
    int rl[8];
    #pragma unroll
    for (int r = 0; r < 8; ++r)
        rl[r] = labels[(rowbase + r + 8 * hf) & (BSZ - 1)];

    // Per-lane online softmax state per row slot. (m=0, l=0) is a valid neutral
    // start: the pair means "sum of exp(v - m)" with reference point m.
    float m[8], ls[8];
    #pragma unroll
    for (int r = 0; r < 8; ++r) { m[r] = 0.0f; ls[r] = 0.0f; }

    if (wave == 0)
        __builtin_amdgcn_s_wait_tensorcnt(0);   // panel 0 landed
    __syncthreads();

    for (int p = 0; p < NPANEL; ++p) {
        const int colbase = p * PANEL;
        float* buf = lds + (p & 1) * PANEL_FLOATS;

        // Issue next panel into the other buffer (compute of p overlaps the DMA).
        if (wave == 0 && p + 1 < NPANEL)
            tdm_load_panel(feat + cf_base(colbase + PANEL),
                           lds_offset_of(lds + ((p + 1) & 1) * PANEL_FLOATS));

        #pragma unroll
        for (int n0 = 0; n0 < PANEL; n0 += 16) {
            v8f acc = {0.f, 0.f, 0.f, 0.f, 0.f, 0.f, 0.f, 0.f};
            const float* bcol = buf + (n0 + l16) * LDS_STRIDE + kc;
            #pragma unroll
            for (int t = 0; t < 32; ++t) {
                v2f b = *(const v2f*)(bcol + 4 * t);
                acc = __builtin_amdgcn_wmma_f32_16x16x4_f32(
                    false, areg[t], false, b, (short)0, acc, false, false);
            }
            // ---- mask + scale + online (max, expsum) update ----
            const int cl = labels[(colbase + n0 + l16) & (BSZ - 1)];
            #pragma unroll
            for (int r = 0; r < 8; ++r) {
                float v  = (rl[r] != cl) ? acc[r] * INV_TEMP : 0.0f;
                float nm = fmaxf(m[r], v);
                ls[r] = ls[r] * __expf(m[r] - nm) + __expf(v - nm);
                m[r]  = nm;
            }
        }
        __syncthreads();                        // everyone done with buf[p&1]
        if (wave == 0 && p + 1 < NPANEL)
            __builtin_amdgcn_s_wait_tensorcnt(0);  // panel p+1 landed
        __syncthreads();                        // publish next buffer
    }

    // ---- Merge the 16 lanes of each half-wave (columns n ≡ lane (mod 16)) ----
    #pragma unroll
    for (int r = 0; r < 8; ++r) {
        float mm = m[r], ll = ls[r];
        #pragma unroll
        for (int s = 1; s < 16; s <<= 1) {
            float mo = __shfl_xor(mm, s, 32);
            float lo = __shfl_xor(ll, s, 32);
            float M  = fmaxf(mm, mo);
            ll = ll * __expf(mm - M) + lo * __expf(mo - M);
            mm = M;
        }
        ls[r] = ll;
    }
    float tot = 0.0f;
    #pragma unroll
    for (int r = 0; r < 8; ++r) tot += ls[r];
    if (l16 == 0) atomicAdd(Sacc, tot);   // lanes 0 and 16: rows 0-7 / 8-15
}

__global__ void finalize_kernel(const int* __restrict__ labels,
                                const unsigned* __restrict__ hist,
                                const float* __restrict__ Sacc,
                                float* __restrict__ out) {
    __shared__ long long sN[256];
    __shared__ int       sV[256];
    long long Nl = 0;
    int V = 0;
    for (int i = threadIdx.x; i < BSZ; i += 256) {
        long long c = (long long)hist[labels[i] & 127];
        Nl += 2LL * ((long long)N_TOT - 2LL * c);  // two tiled rows per base row
        V  += (c < (long long)BSZ) ? 2 : 0;
    }
    sN[threadIdx.x] = Nl;
    sV[threadIdx.x] = V;
    __syncthreads();
    for (int s = 128; s > 0; s >>= 1) {
        if (threadIdx.x < s) {
            sN[threadIdx.x] += sN[threadIdx.x + s];
            sV[threadIdx.x] += sV[threadIdx.x + s];
        }
        __syncthreads();
    }
    if (threadIdx.x == 0) {
        float loss;
        if (sN[0] == 0) {
            loss = logf(EPSF);   // all labels identical: every row invalid, xv = 0
        } else {
            float S  = *Sacc;
            float x  = (float)((double)S / (double)sN[0]);
            int   Vc = sV[0];
            loss = (Vc * logf(x + EPSF) + (N_TOT - Vc) * logf(EPSF)) / (float)N_TOT;
        }
        *out = loss;
    }
}

extern "C" void kernel_launch(void* const* d_in, const int* in_sizes, int n_in,
                              void* d_out, int out_size, void* d_ws, size_t ws_size,
                              hipStream_t stream) {
    const float* feat   = (const float*)d_in[0];  // [4096, 2, 128] fp32
    const int*   labels = (const int*)d_in[1];    // [4096] int32
    float* out = (float*)d_out;

    float*    ws_f = (float*)d_ws;        // ws_f[0] = S accumulator
    unsigned* hist = (unsigned*)d_ws + 1; // 128 histogram bins

    init_ws_kernel<<<1, 256, 0, stream>>>(ws_f);
    hist_kernel<<<BSZ / 256, 256, 0, stream>>>(labels, hist);
    gram_expsum_kernel<<<N_TOT / ROWS_PER_BLOCK, 256,
                         2 * PANEL_BYTES, stream>>>(feat, labels, ws_f);
    finalize_kernel<<<1, 256, 0, stream>>>(labels, hist, ws_f, out);
}